// GNNPolicy_59846074302835
// MI455X (gfx1250) — compile-verified
//
#include <hip/hip_runtime.h>
#include <hip/hip_bf16.h>

// ---------------------------------------------------------------------------
// GNNPolicy forward for MI455X (gfx1250, wave32).
//
// Math notes (vs reference):
//  * LN over the 1-wide edge feature is a constant (= edge_ln_b), so the edge
//    term folds into the fl-bias:  z = FL[tgt] + FR[src] + const.
//  * segment_sum(relu(LN(z)) @ fn.w + fn.b) == segsum(relu(LN(z))) @ fn.w
//    + deg * fn.b  -> the 64x64 GEMM moves from 1M edges to 50-100k nodes.
//  * All dense GEMMs are [N,64]x[64,64] in f32 via V_WMMA_F32_16X16X4_F32
//    (full f32 precision; compute is not the bottleneck - the edge
//    gather/scatter through L2 is).
// ---------------------------------------------------------------------------

typedef float v2f __attribute__((ext_vector_type(2)));
typedef float v8f __attribute__((ext_vector_type(8)));

#define LDSK 66   // padded k-stride for transposed weights in LDS (even -> 8B aligned pairs)
#define EPS  1e-5f

// ------------------------------ zero fill ----------------------------------
__global__ void k_zero(float* __restrict__ p, long n) {
  long i = (long)blockIdx.x * blockDim.x + threadIdx.x;
  long stride = (long)gridDim.x * blockDim.x;
  for (; i < n; i += stride) p[i] = 0.0f;
}

// ------------------------------ degrees ------------------------------------
__global__ void k_degree(const int* __restrict__ tgt, float* __restrict__ deg, int nE) {
  int i = blockIdx.x * blockDim.x + threadIdx.x;
  if (i < nE) atomicAdd(&deg[tgt[i]], 1.0f);
}

// ------------------- small projection: LN(F) -> F x 64 + ReLU --------------
template <int F>
__global__ void k_proj(const float* __restrict__ x, const float* __restrict__ lnw,
                       const float* __restrict__ lnb, const float* __restrict__ w1,
                       const float* __restrict__ b1, float* __restrict__ y, int N) {
  int idx = blockIdx.x * blockDim.x + threadIdx.x;
  int node = idx >> 6, col = idx & 63;
  if (node >= N) return;
  float xv[F];
  float m = 0.0f;
#pragma unroll
  for (int f = 0; f < F; ++f) { xv[f] = x[node * F + f]; m += xv[f]; }
  m *= (1.0f / F);
  float v = 0.0f;
#pragma unroll
  for (int f = 0; f < F; ++f) { float d = xv[f] - m; v += d * d; }
  v *= (1.0f / F);
  float inv = rsqrtf(v + EPS);
  float acc = b1[col];
#pragma unroll
  for (int f = 0; f < F; ++f) {
    float xn = (xv[f] - m) * inv * lnw[f] + lnb[f];
    acc = fmaf(xn, w1[f * 64 + col], acc);
  }
  y[(size_t)node * 64 + col] = fmaxf(acc, 0.0f);
}

// ------------------- generic [N,64] x [64,64] WMMA GEMM --------------------
// Y[r,c] = act( X@W (+ X2@W2) (+ bias[c]) (+ s2*bias2[c]) (+ rowscale[r]*rowbias[c]) )
// One wave per 16-row strip, 4 n-tiles of 16 cols, K=64 in 16 steps of wmma 16x16x4 f32.
template <bool RELU, bool DUAL>
__global__ __launch_bounds__(256) void k_gemm64(
    const float* __restrict__ X, const float* __restrict__ W,
    const float* __restrict__ X2, const float* __restrict__ W2,
    const float* __restrict__ bias, const float* __restrict__ bias2,
    const float* __restrict__ s2ptr, const float* __restrict__ rowscale,
    const float* __restrict__ rowbias, float* __restrict__ Y, int N) {
  __shared__ float lds[(DUAL ? 2 : 1) * LDSK * 64];

  // Cooperatively transpose W (row-major [k][n]) into LDS as Wt[n][k].
  for (int i = threadIdx.x; i < 4096; i += 256) {
    int k = i >> 6, n = i & 63;
    lds[n * LDSK + k] = W[i];
    if (DUAL) lds[LDSK * 64 + n * LDSK + k] = W2[i];
  }
  __syncthreads();

  int wave = threadIdx.x >> 5;
  int lane = threadIdx.x & 31;
  int row0 = (blockIdx.x * 8 + wave) * 16;
  if (row0 >= N) return;  // wave-uniform; EXEC stays all-ones for WMMA

  int m  = lane & 15;
  int hi = (lane >= 16) ? 1 : 0;
  int rowA = row0 + m;
  if (rowA > N - 1) rowA = N - 1;  // clamp tail loads (N here is always /16 anyway)

  v8f acc[4] = {};

  const float* xrow = X + (size_t)rowA * 64;
#pragma unroll
  for (int kk = 0; kk < 16; ++kk) {
    int k0 = 4 * kk + 2 * hi;            // A: lanes0-15 hold K=k0,k0+1; lanes16-31 K=k0+2,k0+3
    v2f a = *(const v2f*)(xrow + k0);
#pragma unroll
    for (int t = 0; t < 4; ++t) {
      int n = 16 * t + m;                 // B: lane = column n, same K split
      v2f b = *(const v2f*)(&lds[n * LDSK + k0]);
      acc[t] = __builtin_amdgcn_wmma_f32_16x16x4_f32(false, a, false, b,
                                                     (short)0, acc[t], false, false);
    }
  }
  if (DUAL) {
    const float* x2row = X2 + (size_t)rowA * 64;
    const float* lds2 = &lds[LDSK * 64];
#pragma unroll
    for (int kk = 0; kk < 16; ++kk) {
      int k0 = 4 * kk + 2 * hi;
      v2f a = *(const v2f*)(x2row + k0);
#pragma unroll
      for (int t = 0; t < 4; ++t) {
        int n = 16 * t + m;
        v2f b = *(const v2f*)(&lds2[n * LDSK + k0]);
        acc[t] = __builtin_amdgcn_wmma_f32_16x16x4_f32(false, a, false, b,
                                                       (short)0, acc[t], false, false);
      }
    }
  }

  // Epilogue. C/D layout: VGPR j: lanes0-15 -> row row0+j, lanes16-31 -> row row0+j+8.
  float s2 = (bias2 && s2ptr) ? s2ptr[0] : 0.0f;
  float bv[4], b2v[4], rbv[4];
#pragma unroll
  for (int t = 0; t < 4; ++t) {
    int col = 16 * t + m;
    bv[t]  = bias    ? bias[col]         : 0.0f;
    b2v[t] = bias2   ? s2 * bias2[col]   : 0.0f;
    rbv[t] = rowbias ? rowbias[col]      : 0.0f;
  }
#pragma unroll
  for (int j = 0; j < 8; ++j) {
    int row = row0 + j + 8 * hi;
    bool ok = row < N;
    int rc = ok ? row : (N - 1);
    float rs = rowscale ? rowscale[rc] : 0.0f;
#pragma unroll
    for (int t = 0; t < 4; ++t) {
      float y = acc[t][j] + bv[t] + b2v[t] + rs * rbv[t];
      if (RELU) y = fmaxf(y, 0.0f);
      if (ok) Y[(size_t)row * 64 + 16 * t + m] = y;
    }
  }
}

// --------- per-edge: z = FLB[tgt]+FR[src]; relu(LN(z)); atomic scatter ------
__global__ __launch_bounds__(256) void k_edge(
    const float* __restrict__ FLB, const float* __restrict__ FR,
    const int* __restrict__ src, const int* __restrict__ tgt,
    const float* __restrict__ lnw, const float* __restrict__ lnb,
    float* __restrict__ S, int nE) {
  int wave = (blockIdx.x * blockDim.x + threadIdx.x) >> 5;
  int lane = threadIdx.x & 31;
  int nwaves = (gridDim.x * blockDim.x) >> 5;
  for (int e = wave; e < nE; e += nwaves) {
    int s = src[e], t = tgt[e];
    v2f za = *(const v2f*)(FLB + (size_t)t * 64 + 2 * lane);
    v2f zb = *(const v2f*)(FR + (size_t)s * 64 + 2 * lane);
    float zx = za.x + zb.x, zy = za.y + zb.y;
    float ssum = zx + zy;
    float ssq = zx * zx + zy * zy;
#pragma unroll
    for (int off = 16; off; off >>= 1) {
      ssum += __shfl_xor(ssum, off, 32);
      ssq  += __shfl_xor(ssq,  off, 32);
    }
    float mean = ssum * (1.0f / 64.0f);
    float var = ssq * (1.0f / 64.0f) - mean * mean;
    float inv = rsqrtf(var + EPS);
    v2f wv = *(const v2f*)(lnw + 2 * lane);
    v2f bv = *(const v2f*)(lnb + 2 * lane);
    float tx = fmaxf((zx - mean) * inv * wv.x + bv.x, 0.0f);
    float ty = fmaxf((zy - mean) * inv * wv.y + bv.y, 0.0f);
    atomicAdd(S + (size_t)t * 64 + 2 * lane, tx);
    atomicAdd(S + (size_t)t * 64 + 2 * lane + 1, ty);
  }
}

// --------------------- in-place LayerNorm over rows of 64 -------------------
__global__ void k_ln64(float* __restrict__ X, const float* __restrict__ w,
                       const float* __restrict__ b, int N) {
  int gw = (blockIdx.x * blockDim.x + threadIdx.x) >> 5;
  int lane = threadIdx.x & 31;
  if (gw >= N) return;
  float* row = X + (size_t)gw * 64;
  v2f z = *(const v2f*)(row + 2 * lane);
  float s = z.x + z.y, sq = z.x * z.x + z.y * z.y;
#pragma unroll
  for (int off = 16; off; off >>= 1) {
    s  += __shfl_xor(s,  off, 32);
    sq += __shfl_xor(sq, off, 32);
  }
  float mean = s * (1.0f / 64.0f);
  float var = sq * (1.0f / 64.0f) - mean * mean;
  float inv = rsqrtf(var + EPS);
  v2f wv = *(const v2f*)(w + 2 * lane);
  v2f bv = *(const v2f*)(b + 2 * lane);
  v2f o;
  o.x = (z.x - mean) * inv * wv.x + bv.x;
  o.y = (z.y - mean) * inv * wv.y + bv.y;
  *(v2f*)(row + 2 * lane) = o;
}

// --------------------------- final dot: [N,64] . w[64] ----------------------
__global__ void k_dot64(const float* __restrict__ X, const float* __restrict__ w,
                        float* __restrict__ out, int N) {
  int gw = (blockIdx.x * blockDim.x + threadIdx.x) >> 5;
  int lane = threadIdx.x & 31;
  if (gw >= N) return;
  v2f x = *(const v2f*)(X + (size_t)gw * 64 + 2 * lane);
  v2f wv = *(const v2f*)(w + 2 * lane);
  float s = x.x * wv.x + x.y * wv.y;
#pragma unroll
  for (int off = 16; off; off >>= 1) s += __shfl_xor(s, off, 32);
  if (lane == 0) out[gw] = s;
}

// ---------------------------------------------------------------------------
static inline int cdiv(long a, long b) { return (int)((a + b - 1) / b); }

extern "C" void kernel_launch(void* const* d_in, const int* in_sizes, int n_in,
                              void* d_out, int out_size, void* d_ws, size_t ws_size,
                              hipStream_t stream) {
  const float* cons_feat = (const float*)d_in[0];
  const int*   eidx      = (const int*)d_in[1];
  // d_in[2] (edge_features) intentionally unused: LN over a 1-wide axis is constant.
  const float* var_feat  = (const float*)d_in[3];

  const int Nc = in_sizes[0] / 4;
  const int nE = in_sizes[1] / 2;
  const int Nv = in_sizes[3] / 6;
  const int* ci = eidx;        // edge_indices[0]
  const int* vi = eidx + nE;   // edge_indices[1]

  // Params: JAX tree-flatten order (dict keys sorted; lists in order; {'b','w'}).
  auto L = [&](int i) { return (const float*)d_in[i]; };
  const float* cons_l1b = L(4);   const float* cons_l1w = L(5);
  const float* cons_l2b = L(6);   const float* cons_l2w = L(7);
  const float* cons_lnb = L(8);   const float* cons_lnw = L(9);
  struct Conv {
    const float *few, *flb, *flw, *fnb, *fnw, *fnlnb, *fnlnw, *frw;
    const float *o1b, *o1w, *o2b, *o2w, *plnb, *plnw;
  } cv[4];
  for (int c = 0; c < 4; ++c) {
    int b = 10 + 14 * c;
    cv[c] = {L(b + 0), L(b + 1), L(b + 2), L(b + 3), L(b + 4), L(b + 5), L(b + 6),
             L(b + 7), L(b + 8), L(b + 9), L(b + 10), L(b + 11), L(b + 12), L(b + 13)};
  }
  const float* edge_lnb = L(66);  // L(67)=edge_ln_w multiplies an exact zero -> unused
  const float* out1b = L(68);     const float* out1w = L(69);
  const float* out2w = L(70);
  const float* var_l1b = L(71);   const float* var_l1w = L(72);
  const float* var_l2b = L(73);   const float* var_l2w = L(74);
  const float* var_lnb = L(75);   const float* var_lnw = L(76);

  // Workspace carve-up (~155 MB of f32).
  float* ws = (float*)d_ws;
  const size_t NV64 = (size_t)Nv * 64, NC64 = (size_t)Nc * 64;
  float* v_a  = ws; ws += NV64;
  float* v_b  = ws; ws += NV64;
  float* c_a  = ws; ws += NC64;
  float* c_b  = ws; ws += NC64;
  float* bufE = ws; ws += NV64;   // FR scratch / H / head tmp
  float* bufF = ws; ws += NV64;   // FLB scratch / proj hidden
  float* bufG = ws; ws += NV64;   // scatter target S -> AGG -> LN (in-place)
  float* deg_c = ws; ws += Nc;
  float* deg_v = ws; ws += Nv;

  auto zero = [&](float* p, long n) {
    k_zero<<<2048, 256, 0, stream>>>(p, n);
  };
  auto gemm = [&](const float* X, const float* W, const float* X2, const float* W2,
                  const float* bias, const float* bias2, const float* s2p,
                  const float* rsc, const float* rb, float* Y, int N,
                  bool relu, bool dual) {
    int blocks = cdiv(N, 128);
    if (dual) {
      if (relu) k_gemm64<true, true><<<blocks, 256, 0, stream>>>(X, W, X2, W2, bias, bias2, s2p, rsc, rb, Y, N);
      else      k_gemm64<false, true><<<blocks, 256, 0, stream>>>(X, W, X2, W2, bias, bias2, s2p, rsc, rb, Y, N);
    } else {
      if (relu) k_gemm64<true, false><<<blocks, 256, 0, stream>>>(X, W, X2, W2, bias, bias2, s2p, rsc, rb, Y, N);
      else      k_gemm64<false, false><<<blocks, 256, 0, stream>>>(X, W, X2, W2, bias, bias2, s2p, rsc, rb, Y, N);
    }
  };

  // Degrees (constant across layers).
  zero(deg_c, Nc);
  zero(deg_v, Nv);
  k_degree<<<cdiv(nE, 256), 256, 0, stream>>>(ci, deg_c, nE);
  k_degree<<<cdiv(nE, 256), 256, 0, stream>>>(vi, deg_v, nE);

  // Projections.
  k_proj<4><<<cdiv((long)Nc * 64, 256), 256, 0, stream>>>(cons_feat, cons_lnw, cons_lnb,
                                                          cons_l1w, cons_l1b, bufF, Nc);
  gemm(bufF, cons_l2w, nullptr, nullptr, cons_l2b, nullptr, nullptr, nullptr, nullptr,
       c_a, Nc, /*relu=*/true, /*dual=*/false);
  k_proj<6><<<cdiv((long)Nv * 64, 256), 256, 0, stream>>>(var_feat, var_lnw, var_lnb,
                                                          var_l1w, var_l1b, bufF, Nv);
  gemm(bufF, var_l2w, nullptr, nullptr, var_l2b, nullptr, nullptr, nullptr, nullptr,
       v_a, Nv, true, false);

  // One half-convolution: right' = conv(left -src/tgt-> right).
  auto conv = [&](const float* left, int Nl, const int* src, const int* tgt,
                  const float* right, int Nr, const float* deg, const Conv& p,
                  float* newright) {
    // FR = left @ fr.w
    gemm(left, p.frw, nullptr, nullptr, nullptr, nullptr, nullptr, nullptr, nullptr,
         bufE, Nl, false, false);
    // FLB = right @ fl.w + fl.b + edge_ln_b*fe.w
    gemm(right, p.flw, nullptr, nullptr, p.flb, p.few, edge_lnb, nullptr, nullptr,
         bufF, Nr, false, false);
    // S = segsum(relu(LN(FLB[tgt]+FR[src])))
    zero(bufG, (long)Nr * 64);
    k_edge<<<cdiv(nE, 8), 256, 0, stream>>>(bufF, bufE, src, tgt, p.fnlnw, p.fnlnb,
                                            bufG, nE);
    // AGG = S @ fn.w + deg*fn.b   (in-place row-wise safe)
    gemm(bufG, p.fnw, nullptr, nullptr, nullptr, nullptr, nullptr, deg, p.fnb,
         bufG, Nr, false, false);
    // LN(post) in place
    k_ln64<<<cdiv(Nr, 8), 256, 0, stream>>>(bufG, p.plnw, p.plnb, Nr);
    // H = relu(LN(AGG) @ o1.w[0:64] + right @ o1.w[64:128] + o1.b)
    gemm(bufG, p.o1w, right, p.o1w + 64 * 64, p.o1b, nullptr, nullptr, nullptr, nullptr,
         bufE, Nr, true, true);
    // right' = H @ o2.w + o2.b
    gemm(bufE, p.o2w, nullptr, nullptr, p.o2b, nullptr, nullptr, nullptr, nullptr,
         newright, Nr, false, false);
  };

  conv(v_a, Nv, vi, ci, c_a, Nc, deg_c, cv[0], c_b);
  conv(c_b, Nc, ci, vi, v_a, Nv, deg_v, cv[1], v_b);
  conv(v_b, Nv, vi, ci, c_b, Nc, deg_c, cv[2], c_a);
  conv(c_a, Nc, ci, vi, v_b, Nv, deg_v, cv[3], v_a);

  // Head: relu(v @ out1 + b) @ out2.
  gemm(v_a, out1w, nullptr, nullptr, out1b, nullptr, nullptr, nullptr, nullptr,
       bufE, Nv, true, false);
  k_dot64<<<cdiv(Nv, 8), 256, 0, stream>>>(bufE, out2w, (float*)d_out, Nv);

  (void)n_in; (void)out_size; (void)ws_size;
}